// Allatom2Allatom_42356967473476
// MI455X (gfx1250) — compile-verified
//
#include <hip/hip_runtime.h>
#include <hip/hip_bf16.h>
#include <math.h>
#include <stdint.h>

// ---- problem constants (from reference) ----
#define D0      64
#define KC      3
#define EDGEC   32
#define NH      4
#define DHEAD   16
#define NRBF    16
#define MAXB    4
#define TOPK    24
#define KIN_DIM 99     // D0 + EDGE + KC
#define KIN_PAD 112    // padded to multiple of 16 (K-steps of 4)
#define KIN_NS  (KIN_PAD / 4)
#define QIN_DIM 67     // D0 + KC
#define QIN_PAD 80
#define EPSF    1e-5f
#define BIGF    1e9f

typedef float v2f __attribute__((ext_vector_type(2)));
typedef float v8f __attribute__((ext_vector_type(8)));

// Fragment layout per CDNA5 ISA 7.12.2 (fp32 16x16x4):
//   A: lane<16 holds A[m][kk], A[m][kk+1]; lane>=16 holds A[m][kk+2], A[m][kk+3]
//   B: mirrored (lane halves carry K rows {kk,kk+1} / {kk+2,kk+3}, lanes = N)
//   D: vgpr r -> row r (lanes 0-15) / row r+8 (lanes 16-31), col = lane&15

// WMMA step with B fragment already resident in registers.
__device__ __forceinline__ v8f wmma_ab(const float* A_, int lda, int kk,
                                       v2f b, int lane, v8f acc) {
  const int hf = lane >> 4;
  const int mn = lane & 15;
  const int k0 = kk + 2 * hf;
  v2f a;
  a.x = A_[mn * lda + k0];
  a.y = A_[mn * lda + k0 + 1];
  return __builtin_amdgcn_wmma_f32_16x16x4_f32(false, a, false, b, (short)0, acc,
                                               false, false);
}

// WMMA step with B read from global weights (single-use matmuls: q, out0).
__device__ __forceinline__ v8f wmma_f32_k4(const float* A_, int lda, int kk,
                                           const float* __restrict__ B_, int ldb,
                                           int brows, int colbase, int lane, v8f acc) {
  const int hf = lane >> 4;
  const int mn = lane & 15;
  const int k0 = kk + 2 * hf;
  v2f a, b;
  a.x = A_[mn * lda + k0];
  a.y = A_[mn * lda + k0 + 1];
  b.x = (k0     < brows) ? B_[(size_t)k0       * ldb + colbase + mn] : 0.0f;
  b.y = (k0 + 1 < brows) ? B_[(size_t)(k0 + 1) * ldb + colbase + mn] : 0.0f;
  return __builtin_amdgcn_wmma_f32_16x16x4_f32(false, a, false, b, (short)0, acc,
                                               false, false);
}

// Hoist loop-invariant weight B-fragments into registers (per wave / col tile).
template <int NS>
__device__ __forceinline__ void load_bfrags(const float* __restrict__ B_, int ldb,
                                            int brows, int colbase, int lane,
                                            v2f* fr) {
  const int hf = lane >> 4, n = lane & 15;
#pragma unroll
  for (int s = 0; s < NS; ++s) {
    const int k0 = 4 * s + 2 * hf;
    v2f b;
    b.x = (k0     < brows) ? B_[(size_t)k0       * ldb + colbase + n] : 0.0f;
    b.y = (k0 + 1 < brows) ? B_[(size_t)(k0 + 1) * ldb + colbase + n] : 0.0f;
    fr[s] = b;
  }
}

// ---------------------------------------------------------------------------
// Kernel 1: brute-force top-K nearest masked neighbors (dense relaxation of
// make_atom_graph). idx = -1 encodes edge_ok == false.
// ---------------------------------------------------------------------------
__global__ __launch_bounds__(256)
void topk_kernel(const int* __restrict__ seq, const float* __restrict__ xyz,
                 const int* __restrict__ aamask, int* __restrict__ idx_out,
                 int M, int A) {
  __shared__ float sx[256], sy[256], sz[256];
  __shared__ int   sm[256];
  const int  i   = blockIdx.x * 256 + threadIdx.x;
  const bool inr = (i < M);
  int mi = 0; float xi = 0.f, yi = 0.f, zi = 0.f;
  if (inr) {
    mi = aamask[seq[i / A] * A + (i % A)];
    xi = xyz[i * 3 + 0]; yi = xyz[i * 3 + 1]; zi = xyz[i * 3 + 2];
  }
  float bd[TOPK]; int bj[TOPK];
#pragma unroll
  for (int k = 0; k < TOPK; ++k) { bd[k] = BIGF; bj[k] = -1; }
  float worst = BIGF; int wp = 0;

  for (int jt = 0; jt < M; jt += 256) {
    __syncthreads();
    const int j = jt + (int)threadIdx.x;
    if (j < M) {
      sm[threadIdx.x] = aamask[seq[j / A] * A + (j % A)];
      sx[threadIdx.x] = xyz[j * 3 + 0];
      sy[threadIdx.x] = xyz[j * 3 + 1];
      sz[threadIdx.x] = xyz[j * 3 + 2];
    } else {
      sm[threadIdx.x] = 0;
    }
    __syncthreads();
    if (inr && mi) {
      for (int t = 0; t < 256; ++t) {
        const int j2 = jt + t;
        if (!sm[t] || j2 == i) continue;
        const float dx = sx[t] - xi, dy = sy[t] - yi, dz = sz[t] - zi;
        const float d2 = dx * dx + dy * dy + dz * dz;
        if (d2 < worst) {
          bd[wp] = d2; bj[wp] = j2;
          worst = bd[0]; wp = 0;
#pragma unroll
          for (int q = 1; q < TOPK; ++q)
            if (bd[q] > worst) { worst = bd[q]; wp = q; }
        }
      }
    }
  }
  if (inr) {
    for (int k = 0; k < TOPK; ++k)
      idx_out[(size_t)i * TOPK + k] = (bd[k] < BIGF * 0.5f) ? bj[k] : -1;
  }
}

// ---------------------------------------------------------------------------
// Kernel 2: fused equivariant attention. One 128-thread (4 wave32) block per
// 16 atoms; wave w owns head w (one 16-col WMMA tile, DH == 16).
// ---------------------------------------------------------------------------
__global__ __launch_bounds__(128)
void attn_kernel(const int* __restrict__ seq,
                 const float* __restrict__ xyz,
                 const int* __restrict__ aamask,
                 const int* __restrict__ num_bonds,
                 const float* __restrict__ state,
                 const float* __restrict__ grads,
                 const float* __restrict__ We,  const float* __restrict__ be,
                 const float* __restrict__ Wq,  const float* __restrict__ Wk,
                 const float* __restrict__ Wv0, const float* __restrict__ Wv1,
                 const float* __restrict__ Wo0, const float* __restrict__ Wself,
                 const float* __restrict__ b0,
                 const int* __restrict__ nbr_idx,
                 float* __restrict__ xyz_out, float* __restrict__ state_out,
                 int M, int A) {
  __shared__ float s_kin[16][KIN_PAD];   // per-neighbor-slot kin rows
  __shared__ float s_qin[16][QIN_PAD];
  __shared__ float s_q[16][D0];
  __shared__ float s_node0[16][D0];
  __shared__ float s_o0[16][D0];
  __shared__ float s_ktile[16][D0];
  __shared__ float s_attn[16][NH][TOPK]; // logits -> attn in place
  __shared__ float s_dirv[16][TOPK][3];
  __shared__ float s_dist[16][TOPK];
  __shared__ float s_ef[16][NRBF];       // per-edge RBF values
  __shared__ int   s_bond[16];           // per-edge bond one-hot index
  __shared__ float s_v1t[16][16];
  __shared__ float s_coef[16][4];
  __shared__ int   s_idx[16][TOPK];
  __shared__ int   s_maski[16];
  __shared__ float s_xi[16][3];

  const int tid  = threadIdx.x;
  const int lane = tid & 31;
  const int wv   = tid >> 5;   // wave id == head id == col-tile id
  const int i0   = blockIdx.x * 16;

  if (tid == 0) { __builtin_prefetch(Wk, 0, 0); __builtin_prefetch(Wv0, 0, 0); }

  // ---- setup: per-atom data ----
  for (int u = tid; u < 16 * TOPK; u += 128)
    s_idx[u / TOPK][u % TOPK] =
        nbr_idx[(size_t)(i0 + u / TOPK) * TOPK + (u % TOPK)];
  for (int u = tid; u < 16 * D0; u += 128) {
    const int a_ = u >> 6, c = u & 63;
    const float v = state[(size_t)(i0 + a_) * D0 + c];
    s_node0[a_][c] = v;
    s_qin[a_][c]   = v;
  }
  if (tid < 16) {
    const int gi = i0 + tid;
    const int s  = seq[gi / A];
    s_maski[tid] = aamask[s * A + (gi % A)];
    s_xi[tid][0] = xyz[gi * 3 + 0];
    s_xi[tid][1] = xyz[gi * 3 + 1];
    s_xi[tid][2] = xyz[gi * 3 + 2];
#pragma unroll
    for (int c = 0; c < KC; ++c) {  // l1 norms
      const float* g = grads + ((size_t)c * M + gi) * 3;
      s_qin[tid][D0 + c] = sqrtf(g[0]*g[0] + g[1]*g[1] + g[2]*g[2] + EPSF);
    }
    for (int p = QIN_DIM; p < QIN_PAD; ++p) s_qin[tid][p] = 0.0f;
  }
  __syncthreads();

  // ---- q = qin @ Wq (per-head 16-col tile per wave) ----
  {
    v8f acc = {};
    for (int kk = 0; kk < QIN_PAD; kk += 4)
      acc = wmma_f32_k4(&s_qin[0][0], QIN_PAD, kk, Wq, 64, QIN_DIM, wv * 16, lane, acc);
    const int hf = lane >> 4, n = lane & 15;
#pragma unroll
    for (int r = 0; r < 8; ++r) s_q[r + 8 * hf][wv * 16 + n] = acc[r];
  }
  __syncthreads();

  // Build kin rows (16 atoms x one neighbor slot) in LDS.
  // Neighbor node0 gather uses the CDNA5 async global->LDS path (ASYNCcnt).
  auto build_kin = [&](int k, bool first_pass) {
    for (int u = tid; u < 16 * 16; u += 128) {   // 16 edges x 16 float4 chunks
      const int a_ = u >> 4, c4 = (u & 15) * 4;
      const int j  = s_idx[a_][k];
      const int jc = (j >= 0) ? j : 0;   // garbage rows get attn == 0 anyway
      const float*   gsrc   = state + (size_t)jc * D0 + c4;
      const uint32_t ldsoff = (uint32_t)(size_t)(&s_kin[a_][c4]);
      asm volatile("global_load_async_to_lds_b128 %0, %1, off"
                   :: "v"(ldsoff), "v"((uint64_t)(size_t)gsrc)
                   : "memory");
    }
    if (tid < 16) {
      const int a_ = tid;
      const int gi = i0 + a_;
      const int j  = s_idx[a_][k];
      const int jc = (j >= 0) ? j : 0;
      float dvx, dvy, dvz, dist;
      if (first_pass) {
        const float rx = xyz[jc * 3 + 0] - s_xi[a_][0];
        const float ry = xyz[jc * 3 + 1] - s_xi[a_][1];
        const float rz = xyz[jc * 3 + 2] - s_xi[a_][2];
        dist = sqrtf(rx * rx + ry * ry + rz * rz + EPSF);
        const float inv_d = 1.0f / dist;
        dvx = rx * inv_d; dvy = ry * inv_d; dvz = rz * inv_d;
        s_dirv[a_][k][0] = dvx; s_dirv[a_][k][1] = dvy; s_dirv[a_][k][2] = dvz;
        s_dist[a_][k] = dist;
      } else {
        dvx = s_dirv[a_][k][0]; dvy = s_dirv[a_][k][1]; dvz = s_dirv[a_][k][2];
        dist = s_dist[a_][k];
      }
      // RBF values + bond index staged to LDS; projection spread over block.
#pragma unroll
      for (int r = 0; r < NRBF; ++r) {
        const float dd = dist - (6.0f / 15.0f) * (float)r;
        s_ef[a_][r] = __expf(-2.0f * dd * dd);    // 1/(2*0.5^2) = 2
      }
      const int ri = gi / A, ai = gi % A, rj = jc / A, aj = jc % A;
      int bond = 0;
      if (j >= 0 && rj == ri)
        bond = num_bonds[(size_t)seq[ri] * A * A + ai * A + aj];
      s_bond[a_] = bond < 0 ? 0 : (bond > MAXB ? MAXB : bond);
      // SE(3)-invariant scalars: <l1[j], dirv>
#pragma unroll
      for (int c = 0; c < KC; ++c) {
        const float* g = grads + ((size_t)c * M + jc) * 3;
        s_kin[a_][D0 + EDGEC + c] = g[0] * dvx + g[1] * dvy + g[2] * dvz;
      }
      for (int p = KIN_DIM; p < KIN_PAD; ++p) s_kin[a_][p] = 0.0f;
    }
    asm volatile("s_wait_asynccnt 0x0" ::: "memory");
    __syncthreads();
    // edge features through We + be: [16,32] outputs over all 128 threads
    for (int u = tid; u < 16 * EDGEC; u += 128) {
      const int a_ = u >> 5, c = u & 31;
      float s = be[c] + We[(NRBF + s_bond[a_]) * EDGEC + c];  // one-hot row
#pragma unroll
      for (int r = 0; r < NRBF; ++r) s += s_ef[a_][r] * We[r * EDGEC + c];
      s_kin[a_][D0 + c] = s;
    }
  };

  // ---- pass 1: K projection + logits (Wk fragments hoisted to registers) ----
  {
    v2f bk[KIN_NS];
    load_bfrags<KIN_NS>(Wk, 64, KIN_DIM, wv * 16, lane, bk);
    for (int k = 0; k < TOPK; ++k) {
      build_kin(k, true);
      __syncthreads();
      v8f acc = {};
#pragma unroll
      for (int s = 0; s < KIN_NS; ++s)
        acc = wmma_ab(&s_kin[0][0], KIN_PAD, 4 * s, bk[s], lane, acc);
      {
        const int hf = lane >> 4, n = lane & 15;
#pragma unroll
        for (int r = 0; r < 8; ++r) s_ktile[r + 8 * hf][wv * 16 + n] = acc[r];
      }
      __syncthreads();
      if (tid < 64) {
        const int a_ = tid >> 2, h = tid & 3;
        float dot = 0.0f;
#pragma unroll
        for (int d = 0; d < DHEAD; ++d)
          dot += s_q[a_][h * DHEAD + d] * s_ktile[a_][h * DHEAD + d];
        s_attn[a_][h][k] = (s_idx[a_][k] >= 0) ? dot * 0.25f : -1.0e9f;
      }
      __syncthreads();
    }
  }

  // ---- masked softmax over neighbors ----
  if (tid < 64) {
    const int a_ = tid >> 2, h = tid & 3;
    float mx = -3.0e38f;
    for (int k = 0; k < TOPK; ++k) mx = fmaxf(mx, s_attn[a_][h][k]);
    float se = 0.0f;
    for (int k = 0; k < TOPK; ++k) {
      const float e = __expf(s_attn[a_][h][k] - mx);
      s_attn[a_][h][k] = e;
      se += e;
    }
    const float inv = 1.0f / se;  // se >= 1 always (max term -> exp(0))
    for (int k = 0; k < TOPK; ++k) s_attn[a_][h][k] *= inv;
  }
  __syncthreads();

  // ---- pass 2: V0/V1 projections, attention-weighted accumulation ----
  v8f o0acc = {};
  float o1x = 0.f, o1y = 0.f, o1z = 0.f;
  {
    v2f bv0[KIN_NS], bv1[KIN_NS];
    load_bfrags<KIN_NS>(Wv0, 64, KIN_DIM, wv * 16, lane, bv0);
    if (wv == 0) load_bfrags<KIN_NS>(Wv1, NH * (1 + KC), KIN_DIM, 0, lane, bv1);
    for (int k = 0; k < TOPK; ++k) {
      build_kin(k, false);
      __syncthreads();
      v8f acc = {};
#pragma unroll
      for (int s = 0; s < KIN_NS; ++s)
        acc = wmma_ab(&s_kin[0][0], KIN_PAD, 4 * s, bv0[s], lane, acc);
      {
        const int hf = lane >> 4;
#pragma unroll
        for (int r = 0; r < 8; ++r) {
          const int a_ = r + 8 * hf;
          o0acc[r] += s_attn[a_][wv][k] * acc[r];
        }
      }
      if (wv == 0) {  // V1 tile: cols = h*4 + c  (H=4, 1+KC=4)
        v8f acc1 = {};
#pragma unroll
        for (int s = 0; s < KIN_NS; ++s)
          acc1 = wmma_ab(&s_kin[0][0], KIN_PAD, 4 * s, bv1[s], lane, acc1);
        const int hf = lane >> 4, col = lane & 15, h = col >> 2;
#pragma unroll
        for (int r = 0; r < 8; ++r) {
          const int a_ = r + 8 * hf;
          s_v1t[a_][col] = s_attn[a_][h][k] * acc1[r];
        }
      }
      __syncthreads();
      if (tid < 64) {  // coef[a][c] = sum_h attn*v1
        const int a_ = tid >> 2, c = tid & 3;
        float s = 0.0f;
#pragma unroll
        for (int h = 0; h < NH; ++h) s += s_v1t[a_][h * 4 + c];
        s_coef[a_][c] = s;
      }
      __syncthreads();
      if (tid < 16) {  // vmsg accumulation into out1
        const int j = s_idx[tid][k];
        if (j >= 0) {
          const float c0 = s_coef[tid][0];
          float vx = c0 * s_dirv[tid][k][0];
          float vy = c0 * s_dirv[tid][k][1];
          float vz = c0 * s_dirv[tid][k][2];
#pragma unroll
          for (int c = 0; c < KC; ++c) {
            const float* g  = grads + ((size_t)c * M + j) * 3;
            const float  cc = s_coef[tid][c + 1];
            vx += cc * g[0]; vy += cc * g[1]; vz += cc * g[2];
          }
          o1x += vx; o1y += vy; o1z += vz;
        }
      }
      __syncthreads();
    }
  }

  // ---- out0 = o0 @ Wo0 + node0 @ Wself + b0 ----
  {
    const int hf = lane >> 4, n = lane & 15;
#pragma unroll
    for (int r = 0; r < 8; ++r) s_o0[r + 8 * hf][wv * 16 + n] = o0acc[r];
  }
  __syncthreads();
  {
    v8f acc = {};
    for (int kk = 0; kk < D0; kk += 4)
      acc = wmma_f32_k4(&s_o0[0][0], D0, kk, Wo0, 64, D0, wv * 16, lane, acc);
    for (int kk = 0; kk < D0; kk += 4)
      acc = wmma_f32_k4(&s_node0[0][0], D0, kk, Wself, 64, D0, wv * 16, lane, acc);
    const int hf = lane >> 4, n = lane & 15;
    const int col = wv * 16 + n;
    const float bb = b0[col];
#pragma unroll
    for (int r = 0; r < 8; ++r) {
      const int a_ = r + 8 * hf;
      const int gi = i0 + a_;
      const float val = s_maski[a_] ? (acc[r] + bb) : s_node0[a_][col];
      state_out[(size_t)gi * D0 + col] = val;
    }
  }
  // ---- xyz update ----
  if (tid < 16) {
    const int gi = i0 + tid;
    const bool m = s_maski[tid] != 0;
    xyz_out[gi * 3 + 0] = m ? (s_xi[tid][0] + o1x * 0.01f) : s_xi[tid][0];
    xyz_out[gi * 3 + 1] = m ? (s_xi[tid][1] + o1y * 0.01f) : s_xi[tid][1];
    xyz_out[gi * 3 + 2] = m ? (s_xi[tid][2] + o1z * 0.01f) : s_xi[tid][2];
  }
}

// ---------------------------------------------------------------------------
extern "C" void kernel_launch(void* const* d_in, const int* in_sizes, int n_in,
                              void* d_out, int out_size, void* d_ws, size_t ws_size,
                              hipStream_t stream) {
  (void)n_in; (void)out_size; (void)ws_size;
  const int*   seq       = (const int*)d_in[0];
  const float* xyz       = (const float*)d_in[1];
  const int*   aamask    = (const int*)d_in[2];
  const int*   num_bonds = (const int*)d_in[3];
  const float* state     = (const float*)d_in[4];
  const float* grads     = (const float*)d_in[5];
  // d_in[6] == top_k (device scalar); reference uses 24 == TOPK.
  const float* We    = (const float*)d_in[7];
  const float* be    = (const float*)d_in[8];
  const float* Wq    = (const float*)d_in[9];
  const float* Wk    = (const float*)d_in[10];
  const float* Wv0   = (const float*)d_in[11];
  const float* Wv1   = (const float*)d_in[12];
  const float* Wo0   = (const float*)d_in[13];
  const float* Wself = (const float*)d_in[14];
  const float* b0    = (const float*)d_in[15];

  const int M = in_sizes[1] / 3;              // B*L*A
  const int A = in_sizes[3] / in_sizes[2];    // (NAA*A*A)/(NAA*A)

  float* xyz_out   = (float*)d_out;                       // [M,3]
  float* state_out = (float*)d_out + (size_t)M * 3;       // [M,64]
  int*   idx_ws    = (int*)d_ws;                          // [M,24]

  topk_kernel<<<(M + 255) / 256, 256, 0, stream>>>(seq, xyz, aamask, idx_ws, M, A);
  attn_kernel<<<M / 16, 128, 0, stream>>>(seq, xyz, aamask, num_bonds, state, grads,
                                          We, be, Wq, Wk, Wv0, Wv1, Wo0, Wself, b0,
                                          idx_ws, xyz_out, state_out, M, A);
}